// FractalHilbertTokenizer_43112881717568
// MI455X (gfx1250) — compile-verified
//
#include <hip/hip_runtime.h>
#include <hip/hip_bf16.h>
#include <stdint.h>

// FractalHilbertTokenizer for MI455X (gfx1250).
//
// Shapes (fixed by the reference): images (32, 3, 1024, 1024) f32.
// Every leaf is at level 6 -> N = 4096 leaves of 16x16; only the top-left
// 4x4 corner of each is gathered. tokens: (32, 4096, 48) f32, levels: 4096 x i32 == 6.
//
// Roofline: 0 FLOPs; ~100 MB DRAM reads (16B useful per 64B line) + 25 MB
// writes => ~5us at 23.3 TB/s. Pure data movement: no WMMA path is justified.
// We use the CDNA5 async global->LDS engine (ASYNCcnt) for the scattered 16B
// gathers and emit fully coalesced 128b stores from an LDS staging buffer.

#define TOKENS_TOTAL   4096       // 4^6 leaves
#define TOKENS_PER_BLK 64
#define ITEMS_PER_BLK  (TOKENS_PER_BLK * 12)   // 12 = 3 channels * 4 rows
#define THREADS        256                      // 8 wave32
#define TOKEN_FLOATS   48                       // 3 * 4 * 4

// ORDERS = [[0,1,3,2],[2,0,1,3],[1,3,2,0],[3,2,0,1]] packed 2 bits/entry:
// entry (row, col) lives at bit 2*(row*4 + col).
#define ORDERS_PACKED 0x4B2DD2B4u

// Closed-form Hilbert leaf position: base-4 digits of n (MSB first) select
// the child via ORDERS[level % 4]; quadrant q -> dy = q>>1, dx = q&1.
__device__ __forceinline__ void hilbert_yx(int n, int& y_out, int& x_out) {
    int y = 0, x = 0;
#pragma unroll
    for (int l = 0; l < 6; ++l) {
        int d = (n >> (2 * (5 - l))) & 3;                       // visit-order digit
        int q = (ORDERS_PACKED >> ((((l & 3) << 2) + d) << 1)) & 3; // quadrant id
        int half = 512 >> l;                                    // 512,256,...,16
        y += (q >> 1) * half;
        x += (q & 1) * half;
    }
    y_out = y;
    x_out = x;
}

__global__ void __launch_bounds__(THREADS)
fractal_hilbert_tokenize(const float* __restrict__ images,
                         float* __restrict__ tokens) {
    const int b  = blockIdx.y;                 // batch 0..31
    const int n0 = blockIdx.x * TOKENS_PER_BLK;
    const int t  = threadIdx.x;

    // Staging buffer laid out exactly in output order: item i (token-local,
    // channel, row) occupies floats [4i, 4i+4). 64 tokens * 48 floats = 12 KB.
    __shared__ float smem[TOKENS_PER_BLK * TOKEN_FLOATS];

    // ---- Phase 1: async gather, global -> LDS (no VGPR data movement) ----
#pragma unroll
    for (int k = 0; k < 3; ++k) {
        const int i       = t + THREADS * k;       // item in [0, 768)
        const int n_local = i / 12;
        const int rem     = i % 12;
        const int c       = rem >> 2;              // channel 0..2
        const int r       = rem & 3;               // patch row 0..3
        int y, x;
        hilbert_yx(n0 + n_local, y, x);

        // images[b][c][y+r][x .. x+3]  (16 bytes, 16B-aligned: x % 16 == 0)
        const uint64_t gaddr = (uint64_t)(uintptr_t)(images
            + (((size_t)(b * 3 + c)) << 20)        // channel plane: 1024*1024
            + ((size_t)(y + r) << 10)              // row: 1024
            + (size_t)x);

        // Wave-relative LDS byte offset = low 32 bits of the generic pointer;
        // hardware adds LDS_BASE.
        const uint32_t lds_off = (uint32_t)(uintptr_t)(&smem[i * 4]);

        asm volatile("global_load_async_to_lds_b128 %0, %1, off"
                     :
                     : "v"(lds_off), "v"(gaddr)
                     : "memory");
    }

    // Drain this wave's async queue, then sync the workgroup so every lane
    // may read LDS written by other waves' async ops.
    asm volatile("s_wait_asynccnt 0" ::: "memory");
    __syncthreads();

    // ---- Phase 2: coalesced linear write-back, LDS -> global ----
    // Block writes 64*48 = 3072 contiguous floats; each thread stores
    // 3 float4s at stride 256*16B -> fully coalesced global_store_b128.
    float* dst = tokens + ((size_t)b * TOKENS_TOTAL + (size_t)n0) * TOKEN_FLOATS;
#pragma unroll
    for (int k = 0; k < 3; ++k) {
        const int i = t + THREADS * k;
        const float4 v = *reinterpret_cast<const float4*>(&smem[i * 4]);
        *reinterpret_cast<float4*>(dst + (size_t)i * 4) = v;
    }
}

// Second output: levels[n] == 6 for all 4096 leaves (every leaf is at
// MAX_LEVEL). Written as raw int32 bit patterns after the token block.
__global__ void fractal_hilbert_levels(int* __restrict__ levels) {
    const int i = blockIdx.x * blockDim.x + threadIdx.x;
    if (i < TOKENS_TOTAL) levels[i] = 6;
}

extern "C" void kernel_launch(void* const* d_in, const int* in_sizes, int n_in,
                              void* d_out, int out_size, void* d_ws, size_t ws_size,
                              hipStream_t stream) {
    (void)in_sizes; (void)n_in; (void)out_size; (void)d_ws; (void)ws_size;

    const float* images = (const float*)d_in[0];   // (32, 3, 1024, 1024) f32
    float* tokens = (float*)d_out;                 // (32, 4096, 48) f32

    dim3 grid(TOKENS_TOTAL / TOKENS_PER_BLK, 32);  // (64, 32)
    fractal_hilbert_tokenize<<<grid, THREADS, 0, stream>>>(images, tokens);

    // levels follow the flattened tokens in the concatenated output buffer
    int* levels = (int*)(tokens + (size_t)32 * TOKENS_TOTAL * TOKEN_FLOATS);
    fractal_hilbert_levels<<<TOKENS_TOTAL / THREADS, THREADS, 0, stream>>>(levels);
}